// TTSModel_1511828489026
// MI455X (gfx1250) — compile-verified
//
#include <hip/hip_runtime.h>
#include <hip/hip_bf16.h>
#include <math.h>

// Problem constants (match the reference)
#define BB 32
#define CC 80
#define TX 512
#define TY 2048
#define NEGV (-1e9f)
// logp1 = C * (-0.5*log(2*pi)), x_logs == 0
#define LOGP1_CONST (-73.51508265637381f)

typedef __attribute__((ext_vector_type(2))) float v2f;
typedef __attribute__((ext_vector_type(8))) float v8f;

// ---------------------------------------------------------------------------
// wave32 ballot helper
// ---------------------------------------------------------------------------
__device__ inline unsigned ballot32(bool p) {
#if __has_builtin(__builtin_amdgcn_ballot_w32)
  return __builtin_amdgcn_ballot_w32(p);
#else
  return (unsigned)__ballot(p);
#endif
}

// ---------------------------------------------------------------------------
// gfx1250 async global->LDS support (ASYNCcnt path), guarded so compile can
// never regress: falls back to sync load + global_prefetch_b8.
// Builtin signature (from hipcc diagnostic): (int addrspace(1)*, int
// addrspace(3)*, imm offset, imm cpol) -- printed as (__device__ int*, ...).
// ---------------------------------------------------------------------------
#if defined(__AMDGCN__) && __has_builtin(__builtin_amdgcn_global_load_async_to_lds_b32)
#define HAVE_ASYNC_LDS 1
#else
#define HAVE_ASYNC_LDS 0
#endif

#if HAVE_ASYNC_LDS
#define GLOBAL_AS(p) ((__attribute__((address_space(1))) int*)(p))
#define LDS_AS(p)    ((__attribute__((address_space(3))) int*)(p))

__device__ inline void wait_async_le1() {
#if __has_builtin(__builtin_amdgcn_s_wait_asynccnt)
  __builtin_amdgcn_s_wait_asynccnt(1);
#else
  asm volatile("s_wait_asynccnt 0x1" ::: "memory");
#endif
}
__device__ inline void wait_async_le0() {
#if __has_builtin(__builtin_amdgcn_s_wait_asynccnt)
  __builtin_amdgcn_s_wait_asynccnt(0);
#else
  asm volatile("s_wait_asynccnt 0x0" ::: "memory");
#endif
}
#endif

// ---------------------------------------------------------------------------
// Row-stat kernels:
//   cx[b,t] = logp1 + logp4 = LOGP1 - 0.5*sum_c x_mu^2   (masked t < x_len)
//   cz[b,y] = logp2        =        - 0.5*sum_c z^2      (masked y < y_len)
// ---------------------------------------------------------------------------
__global__ __launch_bounds__(256) void rowstats_x_kernel(
    const float* __restrict__ x_mu, const int* __restrict__ x_length,
    float* __restrict__ cx) {
  int gid = blockIdx.x * 256 + threadIdx.x;   // B*TX threads
  int b = gid / TX, t = gid % TX;
  int xl = x_length[b];
  float s = 0.0f;
  if (t < xl) {
    const float* p = x_mu + (size_t)b * CC * TX + t;
    #pragma unroll 4
    for (int c = 0; c < CC; ++c) { float v = p[(size_t)c * TX]; s += v * v; }
  }
  cx[gid] = LOGP1_CONST - 0.5f * s;
}

__global__ __launch_bounds__(256) void rowstats_z_kernel(
    const float* __restrict__ z, const int* __restrict__ y_length,
    float* __restrict__ cz) {
  int gid = blockIdx.x * 256 + threadIdx.x;   // B*TY threads
  int b = gid / TY, y = gid % TY;
  int yl = y_length[b];
  float s = 0.0f;
  if (y < yl) {
    const float* p = z + (size_t)b * CC * TY + y;
    #pragma unroll 4
    for (int c = 0; c < CC; ++c) { float v = p[(size_t)c * TY]; s += v * v; }
  }
  cz[gid] = -0.5f * s;
}

// ---------------------------------------------------------------------------
// WMMA logp kernel: logpT[b][y][t] = cz[b,y] + cx[b,t] + sum_c z[b,c,y]*x_mu[b,c,t]
// D = A x B with M = y (16), N = t (16), K = c (4 per WMMA, 20 steps for C=80).
// Transposed output layout [B,Ty,Tx] makes each DP column read coalesced.
// One wave per 16x16 tile; 8 waves (sharing the same t-tile) per block.
//
// Fragment layouts per CDNA5 ISA 7.12.2 (32-bit A 16x4 / C-D 16x16):
//   A: lanes 0-15 -> M=lane, vgpr0=K0 vgpr1=K1; lanes 16-31 -> K2,K3
//   B: lanes 0-15 -> N=lane, vgpr0=K0 vgpr1=K1; lanes 16-31 -> K2,K3
//   D: vgpr r, lane L -> (M = r + 8*(L>>4), N = L&15)
// ---------------------------------------------------------------------------
__global__ __launch_bounds__(256) void logp_wmma_kernel(
    const float* __restrict__ x_mu, const float* __restrict__ z,
    const int* __restrict__ x_length, const int* __restrict__ y_length,
    const float* __restrict__ cx, const float* __restrict__ cz,
    float* __restrict__ logpT) {
  const int lane = threadIdx.x & 31;
  const int wave = threadIdx.x >> 5;
  const int tile = blockIdx.x * 8 + wave;        // [0, B*(TX/16)*(TY/16))
  const int ty = tile & ((TY / 16) - 1);         // 0..127
  const int tx = (tile >> 7) & ((TX / 16) - 1);  // 0..31
  const int b  = tile >> 12;                     // 0..31
  const int y0 = ty * 16, t0 = tx * 16;

  const int xl = x_length[b], yl = y_length[b];
  const int half = lane >> 4;
  const int mn   = lane & 15;
  const int yA = y0 + mn;   // A-fragment row  (M)
  const int tB = t0 + mn;   // B-fragment col  (N)
  const float amask = (yA < yl) ? 1.0f : 0.0f;   // z mask
  const float bmask = (tB < xl) ? 1.0f : 0.0f;   // x_mu mask
  const float* zb = z    + (size_t)b * CC * TY;
  const float* xb = x_mu + (size_t)b * CC * TX;

#if defined(__AMDGCN__) && __has_builtin(__builtin_amdgcn_wmma_f32_16x16x4_f32)
  v8f acc = {};
  #pragma unroll
  for (int c0 = 0; c0 < CC; c0 += 4) {
    const int ca = c0 + 2 * half;
    v2f a, bf;
    a.x  = zb[(size_t)ca * TY + yA] * amask;
    a.y  = zb[(size_t)(ca + 1) * TY + yA] * amask;
    bf.x = xb[(size_t)ca * TX + tB] * bmask;
    bf.y = xb[(size_t)(ca + 1) * TX + tB] * bmask;
    acc = __builtin_amdgcn_wmma_f32_16x16x4_f32(
        /*neg_a=*/false, a, /*neg_b=*/false, bf,
        /*c_mod=*/(short)0, acc, /*reuse_a=*/false, /*reuse_b=*/false);
  }
  float d[8];
  #pragma unroll
  for (int r = 0; r < 8; ++r) d[r] = acc[r];
#else
  // Scalar fallback (host pass / builtin unavailable): same fragment semantics.
  float d[8];
  #pragma unroll
  for (int r = 0; r < 8; ++r) {
    const int row = y0 + r + 8 * half;
    const float rm = (row < yl) ? 1.0f : 0.0f;
    float s = 0.0f;
    for (int c = 0; c < CC; ++c)
      s += zb[(size_t)c * TY + row] * rm * xb[(size_t)c * TX + tB] * bmask;
    d[r] = s;
  }
#endif

  const float cxv = cx[b * TX + tB];
  float* outp = logpT + (size_t)b * TY * TX;
  #pragma unroll
  for (int r = 0; r < 8; ++r) {
    const int row = y0 + r + 8 * half;                    // M = y
    outp[(size_t)row * TX + tB] = d[r] + cxv + cz[b * TY + row];
  }
}

// ---------------------------------------------------------------------------
// MAS DP + backtrack. One workgroup (512 threads == Tx) per batch.
//
// LDS layout (dynamic):
//   vbuf  : 2 * TX floats    -- double-buffered DP column (one barrier/step)
//   bits  : TY * TX/32 uints -- backtrack decision bits (ballot-packed)
//   colbuf: 2 * TX floats    -- async-prefetched logp columns (per-lane slots)
//
// Column y lives in vbuf[y&1]; step y reads vbuf[(y-1)&1] and writes
// vbuf[y&1], so a single __syncthreads per step is sufficient.
// Column y+1 of logp is fetched with GLOBAL_LOAD_ASYNC_TO_LDS_B32 while
// column y is consumed (loads complete in order -> s_wait_asynccnt 1).
// ---------------------------------------------------------------------------
__global__ __launch_bounds__(512) void mas_dp_kernel(
    const float* __restrict__ logpT, const int* __restrict__ x_length,
    const int* __restrict__ y_length, int* __restrict__ idx_out) {
  extern __shared__ unsigned char smem[];
  float* vbuf = (float*)smem;                                        // 2*TX
  unsigned* bits = (unsigned*)(smem + 2 * TX * sizeof(float));       // TY*TX/32
  float* colbuf = (float*)(smem + 2 * TX * sizeof(float)
                                + (size_t)TY * (TX / 32) * sizeof(unsigned)); // 2*TX

  const int b = blockIdx.x;
  const int x = threadIdx.x;
  const int xl = x_length[b], yl = y_length[b];
  const float* lp = logpT + (size_t)b * TY * TX;

  float v = NEGV;
  vbuf[TX + x] = NEGV;          // buffer ((0-1)&1)==1 holds virtual column -1
#if HAVE_ASYNC_LDS
  __builtin_amdgcn_global_load_async_to_lds_b32(
      GLOBAL_AS(lp + x), LDS_AS(&colbuf[x]), 0, 0);   // column 0 -> buffer 0
#endif
  __syncthreads();

  for (int y = 0; y < TY; ++y) {
    const int cur = y & 1;
    const int nxt = cur ^ 1;    // also == (y-1)&1

#if HAVE_ASYNC_LDS
    if (y + 1 < TY) {
      __builtin_amdgcn_global_load_async_to_lds_b32(
          GLOBAL_AS(lp + (size_t)(y + 1) * TX + x),
          LDS_AS(&colbuf[nxt * TX + x]), 0, 0);
      wait_async_le1();         // column y resident; y+1 still in flight
    } else {
      wait_async_le0();
    }
    const float l = colbuf[cur * TX + x];   // per-lane slot: no barrier needed
#else
    const float l = lp[(size_t)y * TX + x];
    if (y + 1 < TY) __builtin_prefetch(&lp[(size_t)(y + 1) * TX + x], 0, 3);
#endif

    const float prevv = (x == 0) ? ((y == 0) ? 0.0f : NEGV)
                                 : vbuf[nxt * TX + x - 1];
    const float vcur  = (x == y) ? NEGV : v;
    int lo = xl + y - yl; if (lo < 0) lo = 0;
    int hi = (xl - 1 < y) ? (xl - 1) : y;
    const bool valid = (x >= lo) && (x <= hi) && (y < yl);
    const float vnew = valid ? (fmaxf(vcur, prevv) + l) : NEGV;

    vbuf[cur * TX + x] = vnew;
    v = vnew;
    __syncthreads();            // single barrier per step (double-buffered)

    const bool bit = (x > 0) && (vnew < vbuf[cur * TX + x - 1]);
    const unsigned m = ballot32(bit);
    if ((x & 31) == 0) bits[y * (TX / 32) + (x >> 5)] = m;
  }
  __syncthreads();

  if (x == 0) {   // serial backtrack over LDS bits
    int index = xl - 1;
    for (int y = TY - 1; y >= 0; --y) {
      const bool active = (y < yl);
      idx_out[b * TY + y] = active ? index : -1;
      const int py = (y > 0) ? (y - 1) : 0;
      const unsigned w = bits[py * (TX / 32) + (index >> 5)];
      const bool cmp = (w >> (index & 31)) & 1u;
      const bool move = (index != 0) && ((index == y) || cmp);
      if (active && move) --index;
    }
  }
}

// ---------------------------------------------------------------------------
// Output kernels: one-hot path, duration = log(1e-8 + count)*mask,
// x_mu_aligned = gather of x_mu columns by idx (path is one-hot).
// ---------------------------------------------------------------------------
__global__ __launch_bounds__(256) void path_kernel(
    const int* __restrict__ idx, float* __restrict__ path) {
  size_t gid = (size_t)blockIdx.x * 256 + threadIdx.x;   // B*TX*TY
  const int y = (int)(gid % TY);
  const int x = (int)((gid / TY) % TX);
  const int b = (int)(gid / ((size_t)TX * TY));
  path[gid] = (idx[b * TY + y] == x) ? 1.0f : 0.0f;
}

__global__ __launch_bounds__(512) void duration_kernel(
    const int* __restrict__ idx, const int* __restrict__ x_length,
    float* __restrict__ dur) {
  __shared__ int sidx[TY];
  const int b = blockIdx.x;
  for (int i = threadIdx.x; i < TY; i += 512) sidx[i] = idx[b * TY + i];
  __syncthreads();
  const int x = threadIdx.x;
  int cnt = 0;
  #pragma unroll 8
  for (int y = 0; y < TY; ++y) cnt += (sidx[y] == x) ? 1 : 0;
  const float d = logf(1e-8f + (float)cnt);
  dur[b * TX + x] = (x < x_length[b]) ? d : 0.0f;
}

__global__ __launch_bounds__(256) void align_kernel(
    const float* __restrict__ x_mu, const int* __restrict__ idx,
    float* __restrict__ out) {
  size_t gid = (size_t)blockIdx.x * 256 + threadIdx.x;   // B*C*TY
  const int y = (int)(gid % TY);
  const int c = (int)((gid / TY) % CC);
  const int b = (int)(gid / ((size_t)CC * TY));
  const int t = idx[b * TY + y];
  out[gid] = (t >= 0) ? x_mu[((size_t)b * CC + c) * TX + t] : 0.0f;
}

// ---------------------------------------------------------------------------
// Host launcher
// ---------------------------------------------------------------------------
extern "C" void kernel_launch(void* const* d_in, const int* in_sizes, int n_in,
                              void* d_out, int out_size, void* d_ws, size_t ws_size,
                              hipStream_t stream) {
  const float* x_mu = (const float*)d_in[0];   // [B,C,TX]
  const float* z    = (const float*)d_in[1];   // [B,C,TY]
  const int* x_len  = (const int*)d_in[2];     // [B]
  const int* y_len  = (const int*)d_in[3];     // [B]

  float* out = (float*)d_out;
  float* out_aligned = out;                          // [B,C,TY]
  float* out_dur     = out + (size_t)BB * CC * TY;   // [B,1,TX]
  float* out_path    = out_dur + (size_t)BB * TX;    // [B,TX,TY]

  // Workspace partition (all 4-byte aligned)
  float* cx  = (float*)d_ws;                         // B*TX
  float* cz  = cx + (size_t)BB * TX;                 // B*TY
  int*   idx = (int*)(cz + (size_t)BB * TY);         // B*TY

  // logpT scratch [B,TY,TX] aliases the path output region (same element
  // count); path_kernel fully overwrites it afterwards.
  float* logpT = out_path;

  rowstats_x_kernel<<<(BB * TX) / 256, 256, 0, stream>>>(x_mu, x_len, cx);
  rowstats_z_kernel<<<(BB * TY) / 256, 256, 0, stream>>>(z, y_len, cz);

  // B * (TX/16) * (TY/16) = 131072 tiles, 8 waves/block -> 16384 blocks
  logp_wmma_kernel<<<16384, 256, 0, stream>>>(x_mu, z, x_len, y_len, cx, cz, logpT);

  // Dynamic LDS: 2*TX floats (v) + TY*(TX/32) uints (bits) + 2*TX floats (col)
  const size_t dp_lds = 2 * TX * sizeof(float)
                      + (size_t)TY * (TX / 32) * sizeof(unsigned)
                      + 2 * TX * sizeof(float);
  mas_dp_kernel<<<BB, TX, dp_lds, stream>>>(logpT, x_len, y_len, idx);

  path_kernel<<<((size_t)BB * TX * TY) / 256, 256, 0, stream>>>(idx, out_path);
  duration_kernel<<<BB, TX, 0, stream>>>(idx, x_len, out_dur);
  align_kernel<<<((size_t)BB * CC * TY) / 256, 256, 0, stream>>>(x_mu, idx, out_aligned);
}